// FallDetector_2302102471535
// MI455X (gfx1250) — compile-verified
//
#include <hip/hip_runtime.h>

// FallDetector: 1M-step LSTM (INPUT=6, HIDDEN=20) + 4-tag head.
// Kernel 1: x_proj GEMM via V_WMMA_F32_16X16X4_F32 (fp32, bias folded in C).
// Kernel 2: single-wave latency-optimized sequential scan.

#define H  20
#define IN 6
#define G4 80   // 4*H

typedef float v2f __attribute__((ext_vector_type(2)));
typedef float v8f __attribute__((ext_vector_type(8)));

__device__ __forceinline__ float fast_exp(float x) {
  return __builtin_amdgcn_exp2f(x * 1.44269504088896340736f);
}
__device__ __forceinline__ float fast_sigmoid(float x) {
  return __builtin_amdgcn_rcpf(1.0f + fast_exp(-x));
}
__device__ __forceinline__ float fast_tanh(float x) {
#if __has_builtin(__builtin_amdgcn_tanhf)
  return __builtin_amdgcn_tanhf(x);
#else
  float r;
  // V_TANH_F32 is a TRANS op on gfx1250; pad with v_nop for the co-exec hazard.
  asm volatile("v_tanh_f32 %0, %1\n\tv_nop\n\tv_nop" : "=v"(r) : "v"(x));
  return r;
#endif
}

// ---------------------------------------------------------------------------
// Kernel 1: x_proj[t][j][g] = sum_k seq[t][k]*w_ih[g*20+j][k] + b_ih+b_hh
// One wave per block. A = 16 timesteps x K, B = K x 16 outputs, fp32 WMMA.
// A 16x4 layout: lanes 0-15 hold {K0,K1} of row M=lane, lanes 16-31 {K2,K3}.
// B 4x16 layout (C-analogy): VGPR v holds row K=v (lanes 0-15), K=v+2 (16-31).
// D 16x16: VGPR r -> row r + 8*(lane/16), col = lane%16.
// ---------------------------------------------------------------------------
__global__ __launch_bounds__(32) void xproj_wmma_kernel(
    const float* __restrict__ seq, const float* __restrict__ w_ih,
    const float* __restrict__ b_ih, const float* __restrict__ b_hh,
    float* __restrict__ xp, int t0, int ntiles, int tiles_per_block)
{
  const int lane = threadIdx.x & 31;
  const int half = lane >> 4;   // 0: K 0..1 / rows half, 1: K 2..3
  const int l16  = lane & 15;

  v2f   Bm[5][2];
  float bias[5];
  int   goff[5], joff[5];
#pragma unroll
  for (int nt = 0; nt < 5; ++nt) {
    const int n = nt * 16 + l16;       // output column 0..79
    const int kbase = 2 * half;
#pragma unroll
    for (int v = 0; v < 2; ++v) {
      const int k0 = v + kbase;        // 0..3   (K chunk 0)
      const int k1 = 4 + v + kbase;    // 4..7   (K chunk 1, pad >5 with 0)
      Bm[nt][0][v] = w_ih[n * IN + k0];
      Bm[nt][1][v] = (k1 < IN) ? w_ih[n * IN + k1] : 0.0f;
    }
    bias[nt] = b_ih[n] + b_hh[n];
    goff[nt] = n / H;                  // gate index 0..3
    joff[nt] = n % H;                  // hidden index 0..19
  }

  const int tbeg = blockIdx.x * tiles_per_block;
  int tend = tbeg + tiles_per_block;
  if (tend > ntiles) tend = ntiles;

  for (int tt = tbeg; tt < tend; ++tt) {
    const long trow = (long)t0 + (long)tt * 16 + l16;
    const float* srow = seq + trow * IN;
    v2f A0, A1;
    A0[0] = srow[2 * half + 0];
    A0[1] = srow[2 * half + 1];
    A1[0] = half ? 0.0f : srow[4];
    A1[1] = half ? 0.0f : srow[5];

#pragma unroll
    for (int nt = 0; nt < 5; ++nt) {
      v8f acc;
#pragma unroll
      for (int r = 0; r < 8; ++r) acc[r] = bias[nt];  // bias constant per column
      acc = __builtin_amdgcn_wmma_f32_16x16x4_f32(false, A0, false, Bm[nt][0],
                                                  (short)0, acc, false, false);
      acc = __builtin_amdgcn_wmma_f32_16x16x4_f32(false, A1, false, Bm[nt][1],
                                                  (short)0, acc, false, false);
      // scatter to gate-interleaved chunk-local layout (t, j, gate)
      float* dst = xp + (size_t)(tt * 16 + 8 * half) * G4 + joff[nt] * 4 + goff[nt];
#pragma unroll
      for (int r = 0; r < 8; ++r) dst[(size_t)r * G4] = acc[r];
    }
  }
}

// ---------------------------------------------------------------------------
// Kernel 2: single-wave sequential LSTM scan over one chunk.
// Lane j (<20) owns gate pre-activations i/f/g/o for hidden unit j.
// h broadcast via LDS (same-wave DS ordering, no barriers needed).
// ---------------------------------------------------------------------------
__global__ __launch_bounds__(32) void lstm_scan_kernel(
    const float* __restrict__ xp, const float* __restrict__ seq,
    const float* __restrict__ w_ih, const float* __restrict__ w_hh,
    const float* __restrict__ b_ih, const float* __restrict__ b_hh,
    const float* __restrict__ w_tag, const float* __restrict__ b_tag,
    float* __restrict__ state, float* __restrict__ out,
    int t0, int tcount, int first, int last, int fused)
{
  __shared__ __align__(16) float hsh[32];
  const int j = threadIdx.x;
  if (j >= H) return;  // no barriers below -> early exit of lanes 20..31 is safe

  float wi[H], wf[H], wg[H], wo[H];
#pragma unroll
  for (int k = 0; k < H; ++k) {
    wi[k] = w_hh[(j + 0 * H) * H + k];
    wf[k] = w_hh[(j + 1 * H) * H + k];
    wg[k] = w_hh[(j + 2 * H) * H + k];
    wo[k] = w_hh[(j + 3 * H) * H + k];
  }

  float vi[IN], vf[IN], vg[IN], vo[IN];
  float bi = 0.f, bf = 0.f, bg = 0.f, bo = 0.f;
  if (fused) {
#pragma unroll
    for (int k = 0; k < IN; ++k) {
      vi[k] = w_ih[(j + 0 * H) * IN + k];
      vf[k] = w_ih[(j + 1 * H) * IN + k];
      vg[k] = w_ih[(j + 2 * H) * IN + k];
      vo[k] = w_ih[(j + 3 * H) * IN + k];
    }
    bi = b_ih[j + 0 * H] + b_hh[j + 0 * H];
    bf = b_ih[j + 1 * H] + b_hh[j + 1 * H];
    bg = b_ih[j + 2 * H] + b_hh[j + 2 * H];
    bo = b_ih[j + 3 * H] + b_hh[j + 3 * H];
  }

  float h, c;
  if (first) { h = 0.0f; c = 0.0f; }
  else       { h = state[j]; c = state[H + j]; }
  hsh[j] = h;
  __builtin_amdgcn_wave_barrier();

  // depth-4 prefetch ring for the gate-interleaved x_proj stream
  float4 pre[4];
  if (!fused) {
#pragma unroll
    for (int d = 0; d < 4; ++d)
      pre[d] = *(const float4*)(xp + (size_t)d * G4 + j * 4);
  }

#pragma unroll 4
  for (int t = 0; t < tcount; ++t) {
    float ai, af, ag, ao;
    if (!fused) {
      float4 xv = pre[t & 3];
      if (t + 4 < tcount)
        pre[t & 3] = *(const float4*)(xp + (size_t)(t + 4) * G4 + j * 4);
      ai = xv.x; af = xv.y; ag = xv.z; ao = xv.w;
    } else {
      const float* x = seq + (size_t)(t0 + t) * IN;
      ai = bi; af = bf; ag = bg; ao = bo;
#pragma unroll
      for (int k = 0; k < IN; ++k) {
        const float xk = x[k];
        ai += vi[k] * xk; af += vf[k] * xk;
        ag += vg[k] * xk; ao += vo[k] * xk;
      }
    }

    float hv[H];
#pragma unroll
    for (int q = 0; q < 5; ++q) {
      const float4 h4 = *(const float4*)(hsh + 4 * q);
      hv[4 * q + 0] = h4.x; hv[4 * q + 1] = h4.y;
      hv[4 * q + 2] = h4.z; hv[4 * q + 3] = h4.w;
    }

    // 4 gates x 2 partial accumulator chains -> dependent chain ~10 FMAs
    float ai2 = 0.f, af2 = 0.f, ag2 = 0.f, ao2 = 0.f;
#pragma unroll
    for (int k = 0; k < H; k += 2) {
      ai  += wi[k]     * hv[k];     ai2 += wi[k + 1] * hv[k + 1];
      af  += wf[k]     * hv[k];     af2 += wf[k + 1] * hv[k + 1];
      ag  += wg[k]     * hv[k];     ag2 += wg[k + 1] * hv[k + 1];
      ao  += wo[k]     * hv[k];     ao2 += wo[k + 1] * hv[k + 1];
    }
    ai += ai2; af += af2; ag += ag2; ao += ao2;

    const float ig = fast_sigmoid(ai);
    const float fg = fast_sigmoid(af);
    const float gg = fast_tanh(ag);
    const float og = fast_sigmoid(ao);
    c = fg * c + ig * gg;
    h = og * fast_tanh(c);

    __builtin_amdgcn_wave_barrier();  // keep hv loads before this store
    hsh[j] = h;
    __builtin_amdgcn_wave_barrier();  // keep next-iter loads after this store
  }

  if (last) {
    __builtin_amdgcn_wave_barrier();
    hsh[j] = c;                       // reuse LDS for final cell state
    __builtin_amdgcn_wave_barrier();
    if (j < 4) {
      float acc = b_tag[j];
#pragma unroll
      for (int k = 0; k < H; ++k) acc += w_tag[j * H + k] * hsh[k];
      out[j] = acc;
    }
  } else {
    state[j]     = h;
    state[H + j] = c;
  }
}

// ---------------------------------------------------------------------------
// Host launcher: chunk x_proj through the workspace; state (h,c) lives in the
// first 256 bytes of d_ws. Falls back to fused scan if ws is too small.
// ---------------------------------------------------------------------------
extern "C" void kernel_launch(void* const* d_in, const int* in_sizes, int n_in,
                              void* d_out, int out_size, void* d_ws, size_t ws_size,
                              hipStream_t stream) {
  const float* seq   = (const float*)d_in[0];
  const float* w_ih  = (const float*)d_in[1];
  const float* w_hh  = (const float*)d_in[2];
  const float* b_ih  = (const float*)d_in[3];
  const float* b_hh  = (const float*)d_in[4];
  const float* w_tag = (const float*)d_in[5];
  const float* b_tag = (const float*)d_in[6];
  float* out = (float*)d_out;
  const long T = in_sizes[0] / IN;

  float* state = (float*)d_ws;                   // 40 floats used
  float* xp    = (float*)((char*)d_ws + 256);    // chunk buffer

  long maxTc = 0;
  if (ws_size > 768) maxTc = (long)((ws_size - 768) / (G4 * sizeof(float)));
  maxTc = (maxTc / 16) * 16;

  if (maxTc >= 16 && (T % 16) == 0) {
    const long Tc = (maxTc < T) ? maxTc : T;
    for (long t0 = 0; t0 < T; t0 += Tc) {
      const long tc = ((T - t0) < Tc) ? (T - t0) : Tc;     // multiple of 16
      const int ntiles = (int)(tc / 16);
      const int tpb = 8;
      const int blocks = (ntiles + tpb - 1) / tpb;
      xproj_wmma_kernel<<<blocks, 32, 0, stream>>>(seq, w_ih, b_ih, b_hh, xp,
                                                   (int)t0, ntiles, tpb);
      lstm_scan_kernel<<<1, 32, 0, stream>>>(xp, seq, w_ih, w_hh, b_ih, b_hh,
                                             w_tag, b_tag, state, out,
                                             (int)t0, (int)tc,
                                             (int)(t0 == 0), (int)(t0 + tc >= T), 0);
    }
  } else {
    lstm_scan_kernel<<<1, 32, 0, stream>>>(nullptr, seq, w_ih, w_hh, b_ih, b_hh,
                                           w_tag, b_tag, state, out,
                                           0, (int)T, 1, 1, 1);
  }
}